// HyperConnections_13675175871115
// MI455X (gfx1250) — compile-verified
//
#include <hip/hip_runtime.h>
#include <math.h>

#define S4   4
#define TDIM 5          // S + I
#define DDIM 2048
#define LDIM 2048
#define NRED 28         // 4 ssq + 20 alpha dots + 4 beta dots
#define ROWP 2052       // padded LDS row stride (floats): 2052 % 64 == 4 -> conflict-free D stores
#define EPS  1.1920929e-07f   // jnp.finfo(float32).eps

typedef __attribute__((ext_vector_type(2))) float v2f;
typedef __attribute__((ext_vector_type(8))) float v8f;

#if __has_builtin(__builtin_amdgcn_wmma_f32_16x16x4_f32)
#define HAVE_WMMA_F32X4 1
#else
#define HAVE_WMMA_F32X4 0
#endif

__device__ __forceinline__ float wave_sum32(float v) {
#pragma unroll
  for (int m = 16; m >= 1; m >>= 1) v += __shfl_xor(v, m, 32);
  return v;
}

// Fold norm_weight into the dynamic fn matrices, planar layout:
// w6[t*2048 + d] = nw[d]*afn[d][t]  (t=0..4),  w6[5*2048 + d] = nw[d]*bfn[d]
__global__ void hc_prep(const float* __restrict__ nw,
                        const float* __restrict__ afn,
                        const float* __restrict__ bfn,
                        float* __restrict__ w6) {
  int d = blockIdx.x * blockDim.x + threadIdx.x;
  if (d >= DDIM) return;
  float w = nw[d];
#pragma unroll
  for (int t = 0; t < TDIM; ++t) w6[t * DDIM + d] = w * afn[d * TDIM + t];
  w6[TDIM * DDIM + d] = w * bfn[d];
}

__launch_bounds__(256)
__global__ void hc_main(const float* __restrict__ res,
                        const float* __restrict__ w6,
                        const float* __restrict__ SA,   // (4,5)
                        const float* __restrict__ SB,   // (4,1)
                        const float* __restrict__ pas,  // alpha scale
                        const float* __restrict__ pbs,  // beta scale
                        float* __restrict__ out) {
  __shared__ __align__(16) float sx[S4][ROWP];
  __shared__ float sred[8][NRED];
  __shared__ float stot[NRED];
  __shared__ float scoef[16];   // scoef[n*4+sp] = coefficient of x[sp] in out[n]

  const int tid   = threadIdx.x;
  const int token = blockIdx.x;
  const int l     = token & (LDIM - 1);
  const int b     = token >> 11;

  int rowbase[S4];
#pragma unroll
  for (int s = 0; s < S4; ++s)
    rowbase[s] = ((b * S4 + s) * LDIM + l) * DDIM;   // < 2^27, fits int32

  // ---- phase 0: load x (coalesced b128) into registers + LDS ----
  float4 xs[S4][2];
#pragma unroll
  for (int g = 0; g < 2; ++g) {
    const int d0 = g * 1024 + tid * 4;
#pragma unroll
    for (int s = 0; s < S4; ++s) {
      float4 v = *reinterpret_cast<const float4*>(res + rowbase[s] + d0);
      xs[s][g] = v;
      *reinterpret_cast<float4*>(&sx[s][d0]) = v;
    }
  }

  // ---- phase 1: per-thread partial sums (ssq, x.(w*afn), x.(w*bfn)) ----
  float ssq[S4] = {0.f, 0.f, 0.f, 0.f};
  float da[S4][TDIM] = {};
  float db[S4] = {0.f, 0.f, 0.f, 0.f};
#pragma unroll
  for (int g = 0; g < 2; ++g) {
    const int d0 = g * 1024 + tid * 4;
    float4 wv[TDIM];
#pragma unroll
    for (int t = 0; t < TDIM; ++t)
      wv[t] = *reinterpret_cast<const float4*>(w6 + t * DDIM + d0);
    float4 wb = *reinterpret_cast<const float4*>(w6 + TDIM * DDIM + d0);
#pragma unroll
    for (int s = 0; s < S4; ++s) {
      float4 xv = xs[s][g];
      ssq[s] = fmaf(xv.x, xv.x, fmaf(xv.y, xv.y, fmaf(xv.z, xv.z, fmaf(xv.w, xv.w, ssq[s]))));
#pragma unroll
      for (int t = 0; t < TDIM; ++t)
        da[s][t] = fmaf(xv.x, wv[t].x, fmaf(xv.y, wv[t].y,
                   fmaf(xv.z, wv[t].z, fmaf(xv.w, wv[t].w, da[s][t]))));
      db[s] = fmaf(xv.x, wb.x, fmaf(xv.y, wb.y, fmaf(xv.z, wb.z, fmaf(xv.w, wb.w, db[s]))));
    }
  }

  // ---- wave32 butterfly reduce, then cross-wave tree in LDS ----
  const int wave = tid >> 5;
  const int lane = tid & 31;
  {
    float red[NRED];
#pragma unroll
    for (int s = 0; s < S4; ++s) red[s] = ssq[s];
#pragma unroll
    for (int s = 0; s < S4; ++s)
#pragma unroll
      for (int t = 0; t < TDIM; ++t) red[4 + s * TDIM + t] = da[s][t];
#pragma unroll
    for (int s = 0; s < S4; ++s) red[24 + s] = db[s];
#pragma unroll
    for (int i = 0; i < NRED; ++i) {
      float tv = wave_sum32(red[i]);
      if (lane == 0) sred[wave][i] = tv;
    }
  }
  __syncthreads();
  if (tid < NRED) {
    float t = 0.f;
#pragma unroll
    for (int w = 0; w < 8; ++w) t += sred[w][tid];
    stot[tid] = t;
  }
  __syncthreads();

  // ---- tiny coefficient stage: M[n][sp] = alpha[sp][n+1] + beta[n]*alpha[sp][0] ----
  if (tid < 16) {
    const int n = tid >> 2, sp = tid & 3;
    const float as = pas[0], bs = pbs[0];
    const float rstd_sp = rsqrtf(stot[sp] * (1.0f / DDIM) + EPS);
    const float rstd_n  = rsqrtf(stot[n]  * (1.0f / DDIM) + EPS);
    const float a1 = tanhf(stot[4 + sp * TDIM + (n + 1)] * rstd_sp) * as + SA[sp * TDIM + (n + 1)];
    const float a0 = tanhf(stot[4 + sp * TDIM + 0]       * rstd_sp) * as + SA[sp * TDIM + 0];
    const float bn = tanhf(stot[24 + n] * rstd_n) * bs + SB[n];
    scoef[n * 4 + sp] = a1 + bn * a0;
  }
  __syncthreads();

#if HAVE_WMMA_F32X4
  // ---- phase 2: out^T(16x4 chunk) = x^T(16x4) * M^T(4x16) via V_WMMA_F32_16X16X4_F32 ----
  // A layout: lanes 0-15 hold K=0(V0)/K=1(V1); lanes 16-31 hold K=2(V0)/K=3(V1); M=d offset.
  // B layout (mirrors): V0 = K0(lo lanes)/K2(hi lanes), V1 = K1/K3; N = lane%16 (only N<4 used).
  {
    const int lane_lo = lane & 15;
    const int half    = lane >> 4;
    v2f bm;
    bm.x = (lane_lo < 4) ? scoef[lane_lo * 4 + 2 * half + 0] : 0.f;
    bm.y = (lane_lo < 4) ? scoef[lane_lo * 4 + 2 * half + 1] : 0.f;
#pragma unroll
    for (int i = 0; i < 16; ++i) {
      const int d0 = (wave * 16 + i) * 16;
      v2f am;
      am.x = sx[2 * half + 0][d0 + lane_lo];
      am.y = sx[2 * half + 1][d0 + lane_lo];
      v8f acc = {};
      acc = __builtin_amdgcn_wmma_f32_16x16x4_f32(
          /*neg_a=*/false, am, /*neg_b=*/false, bm,
          /*c_mod=*/(short)0, acc, /*reuse_a=*/false, /*reuse_b=*/false);
      // D: VGPR v -> rows M=v (lanes 0-15) and M=v+8 (lanes 16-31), N=lane_lo.
      if (lane_lo < 4) {
        const int base = d0 + half * 8;
#pragma unroll
        for (int v = 0; v < 8; ++v) sx[lane_lo][base + v] = acc[v];
      }
    }
  }
  __syncthreads();
  // ---- phase 3: coalesced b128 stores from LDS ----
#pragma unroll
  for (int g = 0; g < 2; ++g) {
    const int d0 = g * 1024 + tid * 4;
#pragma unroll
    for (int s = 0; s < S4; ++s)
      *reinterpret_cast<float4*>(out + rowbase[s] + d0) =
          *reinterpret_cast<const float4*>(&sx[s][d0]);
  }
#else
  // ---- VALU fallback: 4x4 mix straight from registers ----
#pragma unroll
  for (int g = 0; g < 2; ++g) {
    const int d0 = g * 1024 + tid * 4;
#pragma unroll
    for (int s = 0; s < S4; ++s) {
      const float c0 = scoef[s * 4 + 0], c1 = scoef[s * 4 + 1];
      const float c2 = scoef[s * 4 + 2], c3 = scoef[s * 4 + 3];
      float4 o;
      o.x = fmaf(c0, xs[0][g].x, fmaf(c1, xs[1][g].x, fmaf(c2, xs[2][g].x, c3 * xs[3][g].x)));
      o.y = fmaf(c0, xs[0][g].y, fmaf(c1, xs[1][g].y, fmaf(c2, xs[2][g].y, c3 * xs[3][g].y)));
      o.z = fmaf(c0, xs[0][g].z, fmaf(c1, xs[1][g].z, fmaf(c2, xs[2][g].z, c3 * xs[3][g].z)));
      o.w = fmaf(c0, xs[0][g].w, fmaf(c1, xs[1][g].w, fmaf(c2, xs[2][g].w, c3 * xs[3][g].w)));
      *reinterpret_cast<float4*>(out + rowbase[s] + d0) = o;
    }
  }
#endif
}

extern "C" void kernel_launch(void* const* d_in, const int* in_sizes, int n_in,
                              void* d_out, int out_size, void* d_ws, size_t ws_size,
                              hipStream_t stream) {
  (void)in_sizes; (void)n_in; (void)out_size; (void)ws_size;
  const float* res = (const float*)d_in[0];  // (16,2048,2048)
  const float* nw  = (const float*)d_in[1];  // (2048,)
  const float* SA  = (const float*)d_in[2];  // (4,5)
  const float* SB  = (const float*)d_in[3];  // (4,1)
  const float* afn = (const float*)d_in[4];  // (2048,5)
  const float* as_ = (const float*)d_in[5];  // scalar
  const float* bfn = (const float*)d_in[6];  // (2048,1)
  const float* bs_ = (const float*)d_in[7];  // scalar
  float* outp = (float*)d_out;
  float* w6   = (float*)d_ws;                // 6*2048 floats = 48 KB scratch

  hc_prep<<<DDIM / 256, 256, 0, stream>>>(nw, afn, bfn, w6);
  hc_main<<<4 * LDIM, 256, 0, stream>>>(res, w6, SA, SB, as_, bs_, outp);
}